// FractalFrameGenerator_65395172049479
// MI455X (gfx1250) — compile-verified
//
#include <hip/hip_runtime.h>
#include <hip/hip_bf16.h>
#include <math.h>

// ---------------------------------------------------------------------------
// FractalFrameGenerator on MI455X (gfx1250).
//
// Math: output is constant on 32x32 tiles (splits are exact halves of powers
// of two, depth<=5). tileColor = W(1024x32) * sigmoidColors(32x3), where
// W[tile][leaf] = sum over <=36 tree paths of the product of softmax probs.
//
// K1: 4 blocks x 256 threads. softmax (64 rows of 32) -> top-3 per row ->
//     per-tile path walk accumulating W in LDS -> leaf mix via
//     v_wmma_f32_16x16x4_f32 (f32 WMMA: no precision loss) -> 1024x3 tile
//     colors in workspace.
// K2: broadcast store, 12.6 MB of f32 via float4 (global_store_b128),
//     bandwidth-bound (~0.54us at 23.3 TB/s).
// ---------------------------------------------------------------------------

typedef __attribute__((ext_vector_type(2))) float v2f;
typedef __attribute__((ext_vector_type(8))) float v8f;

#define PROB_TH 1e-3f

__global__ __launch_bounds__(256) void ffg_tilecolors(
    const float* __restrict__ fc,      // (32,3)
    const float* __restrict__ fs,      // (32,2,32)
    const int*   __restrict__ temp,    // scalar
    float*       __restrict__ tileColors)  // (1024,3) out
{
  __shared__ float sp[64 * 32];   // softmax probs, row r = idx*2+side
  __shared__ int   st3[64 * 3];   // top-3 child indices per row
  __shared__ float sc[96];        // sigmoid(fc), (32,3)
  __shared__ float wl[256 * 33];  // per-thread W[leaf], stride-33 padded

  const int tid  = threadIdx.x;
  const float invT = 1.0f / (float)temp[0];

  // ---- Phase A: softmax rows + top-3 (threads 0..63, one row each) ----
  if (tid < 64) {
    const float* row = fs + tid * 32;
    float m = -1e30f;
    for (int j = 0; j < 32; j++) m = fmaxf(m, row[j] * invT);
    float s = 0.f;
    for (int j = 0; j < 32; j++) {
      float e = expf(row[j] * invT - m);
      s += e;
      sp[tid * 32 + j] = e;
    }
    float inv = 1.0f / s;
    float v0 = -1.f, v1 = -1.f, v2 = -1.f;
    int   i0 = 0,    i1 = 0,    i2 = 0;
    for (int j = 0; j < 32; j++) {
      float p = sp[tid * 32 + j] * inv;
      sp[tid * 32 + j] = p;
      if (p > v0)      { v2 = v1; i2 = i1; v1 = v0; i1 = i0; v0 = p; i0 = j; }
      else if (p > v1) { v2 = v1; i2 = i1; v1 = p;  i1 = j; }
      else if (p > v2) { v2 = p;  i2 = j; }
    }
    st3[tid * 3 + 0] = i0;
    st3[tid * 3 + 1] = i1;
    st3[tid * 3 + 2] = i2;
  }
  // ---- Phase B: sigmoid colors ----
  if (tid < 96) sc[tid] = 1.0f / (1.0f + expf(-fc[tid]));
  __syncthreads();

  // ---- Phase C: per-tile path walk -> W[leaf] in LDS ----
  const int tile = blockIdx.x * 256 + tid;     // 0..1023
  const int ty = tile >> 5, tx = tile & 31;    // tile coords (y = ty*32, x = tx*32)
  float* w = &wl[tid * 33];
  for (int i = 0; i < 32; i++) w[i] = 0.f;

  // depth 0: idx=0 (even -> horizontal split), cy=0,cx=0; side = bit4(ty)
  {
    const int r0 = ((ty >> 4) & 1);            // idx0*2 + side0, idx0 = 0
    const float* p0 = &sp[r0 * 32];
    const int*   t0 = &st3[r0 * 3];
    for (int j0 = 0; j0 < 3; j0++) {           // k = 3 at depth 0
      const int c1 = t0[j0];
      const float q1 = p0[c1];
      if (!(q1 > PROB_TH)) continue;
      const int dir1 = c1 & 1;                 // 1 = vertical split
      const int cy1 = 1, cx1 = 0;              // root consumed one ty bit
      const int side1 = dir1 ? ((tx >> (4 - cx1)) & 1) : ((ty >> (4 - cy1)) & 1);
      const int r1 = c1 * 2 + side1;
      const float* p1 = &sp[r1 * 32];
      const int*   t1 = &st3[r1 * 3];
      for (int j1 = 0; j1 < 3; j1++) {         // k = 3 at depth 1
        const int c2 = t1[j1];
        const float q2 = p1[c2];
        if (!(q2 > PROB_TH)) continue;
        const int cy2 = cy1 + (dir1 ^ 1), cx2 = cx1 + dir1;
        const int dir2 = c2 & 1;
        const int side2 = dir2 ? ((tx >> (4 - cx2)) & 1) : ((ty >> (4 - cy2)) & 1);
        const int r2 = c2 * 2 + side2;
        const float* p2 = &sp[r2 * 32];
        const int*   t2 = &st3[r2 * 3];
        const float w2 = q1 * q2;
        for (int j2 = 0; j2 < 2; j2++) {       // k = 2 at depth 2
          const int c3 = t2[j2];
          const float q3 = p2[c3];
          if (!(q3 > PROB_TH)) continue;
          const int cy3 = cy2 + (dir2 ^ 1), cx3 = cx2 + dir2;
          const int dir3 = c3 & 1;
          const int side3 = dir3 ? ((tx >> (4 - cx3)) & 1) : ((ty >> (4 - cy3)) & 1);
          const int r3 = c3 * 2 + side3;
          const float* p3 = &sp[r3 * 32];
          const int*   t3 = &st3[r3 * 3];
          const float w3 = w2 * q3;
          for (int j3 = 0; j3 < 2; j3++) {     // k = 2 at depth 3
            const int c4 = t3[j3];
            const float q4 = p3[c4];
            if (!(q4 > PROB_TH)) continue;
            const int cy4 = cy3 + (dir3 ^ 1), cx4 = cx3 + dir3;
            const int dir4 = c4 & 1;
            const int side4 = dir4 ? ((tx >> (4 - cx4)) & 1) : ((ty >> (4 - cy4)) & 1);
            const int r4 = c4 * 2 + side4;
            const float w4 = w3 * q4;
            // depth 4: k = 1 -> leaf choice
            const int c5 = st3[r4 * 3 + 0];
            const float q5 = sp[r4 * 32 + c5];
            if (q5 > PROB_TH) w[c5] += w4 * q5;
          }
        }
      }
    }
  }
  __syncthreads();

  // ---- Phase D: leaf mix via f32 WMMA: D(16x16) += A(16x4) * B(4x16) ----
  // A = W chunk (tiles x leaves), B = padded colors (leaves x [RGB,0...]).
  // Layouts (ISA 7.12.2, f32): lane halves carry row-halves, VGPR = index
  // within half. A: lanes0-15 K={kb,kb+1} half0, lanes16-31 half1 (kb+=2).
  // B: v0 = row kb, v1 = row kb+1 per lane-half. D: VGPR r = rows r / r+8.
  const int wave = tid >> 5;
  const int lane = tid & 31;
  const int half = lane >> 4;
  const int ln   = lane & 15;
  for (int cc = 0; cc < 2; cc++) {
    const int base = (wave * 2 + cc) * 16;     // local tile base (0..240)
    v8f acc = {};
#pragma unroll
    for (int s = 0; s < 8; s++) {              // K = 32 in 8 steps of 4
      const int kb = 4 * s + 2 * half;
      v2f a, b;
      a.x = wl[(base + ln) * 33 + kb];
      a.y = wl[(base + ln) * 33 + kb + 1];
      b.x = (ln < 3) ? sc[(kb + 0) * 3 + ln] : 0.f;
      b.y = (ln < 3) ? sc[(kb + 1) * 3 + ln] : 0.f;
      acc = __builtin_amdgcn_wmma_f32_16x16x4_f32(
          false, a, false, b, (short)0, acc, false, false);
    }
    const int gbase = blockIdx.x * 256 + base;
#pragma unroll
    for (int r = 0; r < 8; r++) {
      if (ln < 3) tileColors[(gbase + r + 8 * half) * 3 + ln] = acc[r];
    }
  }
}

// ---- K2: broadcast tile colors to the 3x1024x1024 image (float4 stores) ----
__global__ __launch_bounds__(256) void ffg_write(
    const float* __restrict__ tc, float* __restrict__ out)
{
  const int bid = blockIdx.x;          // 0..3071 : (channel, row)
  const int c = bid >> 10;
  const int y = bid & 1023;
  const int ty = y >> 5;
  const int t = threadIdx.x;
  const int x0 = t << 2;
  const int tx = x0 >> 5;
  const float v = tc[(ty * 32 + tx) * 3 + c];
  float4 o = make_float4(v, v, v, v);
  *(float4*)(out + ((size_t)c << 20) + ((size_t)y << 10) + x0) = o;
}

extern "C" void kernel_launch(void* const* d_in, const int* in_sizes, int n_in,
                              void* d_out, int out_size, void* d_ws, size_t ws_size,
                              hipStream_t stream) {
  (void)in_sizes; (void)n_in; (void)out_size; (void)ws_size;
  const float* fc   = (const float*)d_in[0];   // (32,3)
  const float* fs   = (const float*)d_in[1];   // (32,2,32)
  const int*   temp = (const int*)d_in[2];     // scalar
  float* out = (float*)d_out;                  // (3,1024,1024)
  float* tc  = (float*)d_ws;                   // 1024*3 tile colors

  ffg_tilecolors<<<4, 256, 0, stream>>>(fc, fs, temp, tc);
  ffg_write<<<3 * 1024, 256, 0, stream>>>(tc, out);
}